// TwoGCN_SLClassifier_50208167690317
// MI455X (gfx1250) — compile-verified
//
#include <hip/hip_runtime.h>
#include <hip/hip_bf16.h>

typedef __attribute__((ext_vector_type(2))) float v2f;
typedef __attribute__((ext_vector_type(8))) float v8f;

#define N_NODES 50000
#define FDIM    256
#define BPAIR   16384

// ---------------------------------------------------------------------------
// utility: grid-stride zero
// ---------------------------------------------------------------------------
__global__ void zero_f32(float* __restrict__ p, long n) {
    long i = (long)blockIdx.x * blockDim.x + threadIdx.x;
    long stride = (long)gridDim.x * blockDim.x;
    for (; i < n; i += stride) p[i] = 0.0f;
}

// ---------------------------------------------------------------------------
// degree count: deg[dst] += 1 per edge
// ---------------------------------------------------------------------------
__global__ void deg_kernel(const int* __restrict__ edge, float* __restrict__ deg, int E) {
    int e = blockIdx.x * blockDim.x + threadIdx.x;
    if (e < E) atomicAdd(&deg[edge[E + e]], 1.0f);   // edge_index[1] = dst
}

// dinv = rsqrt(deg + 1), in place
__global__ void dinv_kernel(float* __restrict__ deg, int n) {
    int i = blockIdx.x * blockDim.x + threadIdx.x;
    if (i < n) deg[i] = __frsqrt_rn(deg[i] + 1.0f);
}

// ---------------------------------------------------------------------------
// edge scatter: agg[dst] += h[src] * dinv[src]*dinv[dst]   (4 feats / thread)
// ---------------------------------------------------------------------------
__global__ void scatter_kernel(const float* __restrict__ h, const int* __restrict__ edge,
                               const float* __restrict__ dinv, float* __restrict__ agg, int E) {
    long t = (long)blockIdx.x * blockDim.x + threadIdx.x;
    long total = (long)E * (FDIM / 4);
    if (t >= total) return;
    int e  = (int)(t >> 6);          // FDIM/4 = 64 groups per edge
    int f4 = ((int)t & 63) << 2;
    int src = edge[e];
    int dst = edge[E + e];
    float nrm = dinv[src] * dinv[dst];
    const float* hp = h   + (long)src * FDIM + f4;
    float*       op = agg + (long)dst * FDIM + f4;
    atomicAdd(op + 0, hp[0] * nrm);
    atomicAdd(op + 1, hp[1] * nrm);
    atomicAdd(op + 2, hp[2] * nrm);
    atomicAdd(op + 3, hp[3] * nrm);
}

// ---------------------------------------------------------------------------
// finalize: out = [relu](agg + raw*dinv^2 + bias)
// ---------------------------------------------------------------------------
__global__ void finalize_kernel(const float* __restrict__ agg, const float* __restrict__ raw,
                                const float* __restrict__ dinv, const float* __restrict__ bias,
                                float* __restrict__ out, int relu) {
    long t = (long)blockIdx.x * blockDim.x + threadIdx.x;
    if (t >= (long)N_NODES * FDIM) return;
    int i = (int)(t >> 8);
    int f = (int)t & 255;
    float d = dinv[i];
    float v = agg[t] + raw[t] * d * d + bias[f];
    if (relu) v = fmaxf(v, 0.0f);
    out[t] = v;
}

// ---------------------------------------------------------------------------
// generic fp32 WMMA GEMM: C[M,N] = A[M,K] @ B[K,N] (+bias)(+relu)
// wave32, one 16x16 C tile per wave, V_WMMA_F32_16X16X4_F32, K-step 4.
// A 16x4 layout : lanes 0-15 M=0..15 {K=kk,kk+1}; lanes 16-31 {K=kk+2,kk+3}
// B 4x16 layout : lanes 0-15 N=0..15 {K=kk,kk+1}; lanes 16-31 {K=kk+2,kk+3}
// C layout      : vgpr r -> M = r + 8*half, N = lane&15
// ---------------------------------------------------------------------------
__global__ void gemm_f32_wmma(const float* __restrict__ A, const float* __restrict__ B,
                              const float* __restrict__ bias, float* __restrict__ C,
                              int M, int N, int K, int nTilesN, int relu) {
    int wave = blockIdx.x * (blockDim.x >> 5) + (threadIdx.x >> 5);
    int lane = threadIdx.x & 31;
    int mt = wave / nTilesN;
    int nt = wave - mt * nTilesN;
    if (mt * 16 >= M) return;
    int half = lane >> 4;
    int l    = lane & 15;
    int arow = mt * 16 + l;
    int bcol = nt * 16 + l;

    v8f acc = {};
    for (int kk = 0; kk < K; kk += 4) {
        int ka = kk + half * 2;
        v2f a;
        a.x = A[(long)arow * K + ka];
        a.y = A[(long)arow * K + ka + 1];
        v2f b;
        b.x = B[(long)(ka + 0) * N + bcol];
        b.y = B[(long)(ka + 1) * N + bcol];
        acc = __builtin_amdgcn_wmma_f32_16x16x4_f32(false, a, false, b,
                                                    (short)0, acc, false, false);
    }

    float bv = bias ? bias[bcol] : 0.0f;
#pragma unroll
    for (int r = 0; r < 8; ++r) {
        int row = mt * 16 + half * 8 + r;
        float v = acc[r] + bv;
        if (relu) v = fmaxf(v, 0.0f);
        C[(long)row * N + bcol] = v;
    }
}

// ---------------------------------------------------------------------------
// classifier layer 1 with fused gather/concat:
// z1 = relu(full @ Wc1 + bc1), full[i,k] gathered from
//   k<256   : h2[pair_idx[i,0]][k]
//   k<512   : h2[pair_idx[i,1]][k-256]
//   k<768   : scg[i][k-512]
//   k<3328  : esm[i][k-768]
//   else    : gpt[i][k-3328]
// segment boundaries are multiples of 4 -> each K-step stays in one segment
// ---------------------------------------------------------------------------
__global__ void gemm_pair_wmma(const float* __restrict__ h2, const float* __restrict__ scg,
                               const float* __restrict__ esm, const float* __restrict__ gpt,
                               const int* __restrict__ pidx, const float* __restrict__ W,
                               const float* __restrict__ bias, float* __restrict__ C) {
    const int N = 128, K = 6400, nTilesN = 8, Mt = BPAIR / 16;
    int wave = blockIdx.x * (blockDim.x >> 5) + (threadIdx.x >> 5);
    int lane = threadIdx.x & 31;
    int mt = wave / nTilesN;
    int nt = wave - mt * nTilesN;
    if (mt >= Mt) return;
    int half = lane >> 4;
    int l    = lane & 15;
    int i    = mt * 16 + l;       // pair row
    int bcol = nt * 16 + l;

    int n0 = pidx[2 * i + 0];
    int n1 = pidx[2 * i + 1];

    v8f acc = {};
    for (int kk = 0; kk < K; kk += 4) {
        int ka = kk + half * 2;
        const float* ap;
        if (ka < 256)       ap = h2  + (long)n0 * 256 + ka;
        else if (ka < 512)  ap = h2  + (long)n1 * 256 + (ka - 256);
        else if (ka < 768)  ap = scg + (long)i * 256  + (ka - 512);
        else if (ka < 3328) ap = esm + (long)i * 2560 + (ka - 768);
        else                ap = gpt + (long)i * 3072 + (ka - 3328);
        v2f a;
        a.x = ap[0];
        a.y = ap[1];
        v2f b;
        b.x = W[(long)(ka + 0) * N + bcol];
        b.y = W[(long)(ka + 1) * N + bcol];
        acc = __builtin_amdgcn_wmma_f32_16x16x4_f32(false, a, false, b,
                                                    (short)0, acc, false, false);
    }

    float bv = bias[bcol];
#pragma unroll
    for (int r = 0; r < 8; ++r) {
        int row = mt * 16 + half * 8 + r;
        float v = fmaxf(acc[r] + bv, 0.0f);
        C[(long)row * N + bcol] = v;
    }
}

// ---------------------------------------------------------------------------
// final 64->2 layer (N=2, scalar dot; negligible FLOPs)
// ---------------------------------------------------------------------------
__global__ void final_fc(const float* __restrict__ z2, const float* __restrict__ W,
                         const float* __restrict__ b, float* __restrict__ out) {
    int idx = blockIdx.x * blockDim.x + threadIdx.x;
    if (idx >= BPAIR * 2) return;
    int i = idx >> 1;
    int j = idx & 1;
    float s = b[j];
    const float* z = z2 + (long)i * 64;
#pragma unroll
    for (int k = 0; k < 64; ++k) s += z[k] * W[k * 2 + j];
    out[idx] = s;
}

// ---------------------------------------------------------------------------
// host launcher
// ---------------------------------------------------------------------------
extern "C" void kernel_launch(void* const* d_in, const int* in_sizes, int n_in,
                              void* d_out, int out_size, void* d_ws, size_t ws_size,
                              hipStream_t stream) {
    const float* x    = (const float*)d_in[0];
    const int*   edge = (const int*)d_in[1];
    const float* scg  = (const float*)d_in[2];
    const float* gpt  = (const float*)d_in[3];
    const float* esm  = (const float*)d_in[4];
    const int*   pidx = (const int*)d_in[5];
    const float* W1   = (const float*)d_in[6];
    const float* b1   = (const float*)d_in[7];
    const float* W2   = (const float*)d_in[8];
    const float* b2   = (const float*)d_in[9];
    const float* Wc1  = (const float*)d_in[10];
    const float* bc1  = (const float*)d_in[11];
    const float* Wc2  = (const float*)d_in[12];
    const float* bc2  = (const float*)d_in[13];
    const float* Wc3  = (const float*)d_in[14];
    const float* bc3  = (const float*)d_in[15];
    float* out = (float*)d_out;

    const int E = in_sizes[1] / 2;
    const long nodeElems = (long)N_NODES * FDIM;     // 12.8M floats

    // workspace carve-up (floats)
    float* ws   = (float*)d_ws;
    float* dinv = ws;                    // 50048
    float* bufA = dinv + 50048;          // raw GEMM output
    float* bufB = bufA + nodeElems;      // edge aggregation
    float* bufC = bufB + nodeElems;      // layer output (h1 then h2)
    float* z1   = bufC + nodeElems;      // 16384x128
    float* z2   = z1 + (long)BPAIR * 128;// 16384x64

    const int TB = 256;

    // ---- degree / dinv -------------------------------------------------
    zero_f32<<<256, TB, 0, stream>>>(dinv, 50000);
    deg_kernel<<<(E + TB - 1) / TB, TB, 0, stream>>>(edge, dinv, E);
    dinv_kernel<<<(N_NODES + TB - 1) / TB, TB, 0, stream>>>(dinv, N_NODES);

    // ---- GCN layer 1 ---------------------------------------------------
    {   // bufA = x @ W1   (50000x256x256)
        int tilesM = N_NODES / 16, tilesN = FDIM / 16;
        long waves = (long)tilesM * tilesN;
        gemm_f32_wmma<<<(int)((waves + 7) / 8), TB, 0, stream>>>(
            x, W1, nullptr, bufA, N_NODES, FDIM, FDIM, tilesN, 0);
    }
    zero_f32<<<2048, TB, 0, stream>>>(bufB, nodeElems);
    {
        long total = (long)E * (FDIM / 4);
        scatter_kernel<<<(int)((total + TB - 1) / TB), TB, 0, stream>>>(bufA, edge, dinv, bufB, E);
    }
    finalize_kernel<<<(int)((nodeElems + TB - 1) / TB), TB, 0, stream>>>(
        bufB, bufA, dinv, b1, bufC, 1);

    // ---- GCN layer 2 ---------------------------------------------------
    {   // bufA = h1 @ W2
        int tilesM = N_NODES / 16, tilesN = FDIM / 16;
        long waves = (long)tilesM * tilesN;
        gemm_f32_wmma<<<(int)((waves + 7) / 8), TB, 0, stream>>>(
            bufC, W2, nullptr, bufA, N_NODES, FDIM, FDIM, tilesN, 0);
    }
    zero_f32<<<2048, TB, 0, stream>>>(bufB, nodeElems);
    {
        long total = (long)E * (FDIM / 4);
        scatter_kernel<<<(int)((total + TB - 1) / TB), TB, 0, stream>>>(bufA, edge, dinv, bufB, E);
    }
    finalize_kernel<<<(int)((nodeElems + TB - 1) / TB), TB, 0, stream>>>(
        bufB, bufA, dinv, b2, bufC, 0);   // h2 -> bufC (no relu)

    // ---- classifier: fused gather/concat GEMM (16384 x 6400 x 128) -----
    {
        long waves = (long)(BPAIR / 16) * 8;          // 8192 waves
        gemm_pair_wmma<<<(int)((waves + 7) / 8), TB, 0, stream>>>(
            bufC, scg, esm, gpt, pidx, Wc1, bc1, z1);
    }

    // ---- z2 = relu(z1 @ Wc2 + bc2)  (16384x128x64) ---------------------
    {
        int tilesM = BPAIR / 16, tilesN = 64 / 16;
        long waves = (long)tilesM * tilesN;
        gemm_f32_wmma<<<(int)((waves + 7) / 8), TB, 0, stream>>>(
            z1, Wc2, bc2, z2, BPAIR, 64, 128, tilesN, 1);
    }

    // ---- out = z2 @ Wc3 + bc3  (N=2 scalar) ----------------------------
    final_fc<<<(BPAIR * 2 + TB - 1) / TB, TB, 0, stream>>>(z2, Wc3, bc3, out);
}